// NonGaussConditionedMarkovGP_15479062134815
// MI455X (gfx1250) — compile-verified
//
#include <hip/hip_runtime.h>
#include <hip/hip_bf16.h>

// ---------------------------------------------------------------------------
// Markov-GP (Matern-5/2) filter+smoother on MI455X via temporal parallel scan.
// T = 196608 steps split into 1024 chunks x 192 steps.
//   Phase 0: merge train/test times by binary search (no argsort needed).
//   Phase F: parallel Kalman filter  (SGF associative-scan elements).
//   Phase S: parallel RTS smoother   (backward affine scan).
// CDNA5 paths: TDM tensor_load_to_lds staging (padded, bank-conflict-free),
// s_wait_tensorcnt, v_wmma_f32_16x16x4_f32 batched 4x(4x4) hom products,
// 320KB-LDS single-workgroup Kogge-Stone scans (wave32).
// ---------------------------------------------------------------------------

#define N_TRAIN 131072
#define M_TEST  65536
#define T_TOT   (N_TRAIN + M_TEST)      /* 196608 */
#define NCHUNK  1024
#define CHLEN   192                     /* T_TOT / NCHUNK */
#define CPB     256                     /* chunks per block (4 blocks) */
#define TILE_ELEMS (CPB * CHLEN)        /* 49152 floats per block tile */
#define TILE_LDS_DW (TILE_ELEMS + TILE_ELEMS / 64)  /* +1 DWORD pad per 64 */

// workspace offsets (floats).  total ~5.59M floats (~22.4 MB)
#define OFF_TM    ((size_t)0)
#define OFF_RES   ((size_t)T_TOT)
#define OFF_R     ((size_t)2 * T_TOT)
#define OFF_TIDX  ((size_t)3 * T_TOT)            /* int view */
#define OFF_REC   ((size_t)4 * T_TOT)            /* 24 floats per step */
#define OFF_FAGG  ((size_t)28 * T_TOT)           /* 33 * NCHUNK */
#define OFF_FIN   (OFF_FAGG + (size_t)33 * NCHUNK)  /* 12 * NCHUNK */
#define OFF_SAGG  (OFF_FIN  + (size_t)12 * NCHUNK)  /* 21 * NCHUNK */
#define OFF_SIN   (OFF_SAGG + (size_t)21 * NCHUNK)  /* 12 * NCHUNK */

typedef __attribute__((ext_vector_type(2))) float v2f;
typedef __attribute__((ext_vector_type(8))) float v8f;
typedef __attribute__((ext_vector_type(4))) unsigned int u32x4;
typedef __attribute__((ext_vector_type(8))) int i32x8;
typedef __attribute__((ext_vector_type(4))) int i32x4;

// ------------------------------ 3x3 helpers --------------------------------
__device__ __forceinline__ void mm3(const float* A, const float* B, float* C) {
#pragma unroll
  for (int i = 0; i < 3; ++i)
#pragma unroll
    for (int j = 0; j < 3; ++j)
      C[i*3+j] = A[i*3+0]*B[0*3+j] + A[i*3+1]*B[1*3+j] + A[i*3+2]*B[2*3+j];
}
__device__ __forceinline__ void mm3nt(const float* A, const float* B, float* C) { // A*B^T
#pragma unroll
  for (int i = 0; i < 3; ++i)
#pragma unroll
    for (int j = 0; j < 3; ++j)
      C[i*3+j] = A[i*3+0]*B[j*3+0] + A[i*3+1]*B[j*3+1] + A[i*3+2]*B[j*3+2];
}
__device__ __forceinline__ void mm3tn(const float* A, const float* B, float* C) { // A^T*B
#pragma unroll
  for (int i = 0; i < 3; ++i)
#pragma unroll
    for (int j = 0; j < 3; ++j)
      C[i*3+j] = A[0*3+i]*B[0*3+j] + A[1*3+i]*B[1*3+j] + A[2*3+i]*B[2*3+j];
}
__device__ __forceinline__ void mv3(const float* A, const float* v, float* o) {
#pragma unroll
  for (int i = 0; i < 3; ++i) o[i] = A[i*3+0]*v[0] + A[i*3+1]*v[1] + A[i*3+2]*v[2];
}
__device__ __forceinline__ void inv3(const float* A, float* O) {
  float c00 = A[4]*A[8] - A[5]*A[7];
  float c01 = A[5]*A[6] - A[3]*A[8];
  float c02 = A[3]*A[7] - A[4]*A[6];
  float det = A[0]*c00 + A[1]*c01 + A[2]*c02;
  float id = 1.0f / det;
  O[0] = c00*id; O[1] = (A[2]*A[7]-A[1]*A[8])*id; O[2] = (A[1]*A[5]-A[2]*A[4])*id;
  O[3] = c01*id; O[4] = (A[0]*A[8]-A[2]*A[6])*id; O[5] = (A[2]*A[3]-A[0]*A[5])*id;
  O[6] = c02*id; O[7] = (A[1]*A[6]-A[0]*A[7])*id; O[8] = (A[0]*A[4]-A[1]*A[3])*id;
}

// --------------------------- Matern-5/2 SDE --------------------------------
struct Kern { float lam; float Nm[9]; float Nm2[9]; float Pinf[9]; };

__device__ __forceinline__ Kern make_kern(float variance, float lengthscale) {
  Kern k;
  float lam = sqrtf(5.0f) / lengthscale;
  k.lam = lam;
  float l2 = lam*lam, l3 = l2*lam;
  float F[9] = {0,1,0, 0,0,1, -l3, -3.0f*l2, -3.0f*lam};
#pragma unroll
  for (int i = 0; i < 9; ++i) k.Nm[i] = F[i];
  k.Nm[0] += lam; k.Nm[4] += lam; k.Nm[8] += lam;
  mm3(k.Nm, k.Nm, k.Nm2);
  float kap = l2 / 3.0f;
  float P[9] = {1,0,-kap, 0,kap,0, -kap,0,l2*l2};
#pragma unroll
  for (int i = 0; i < 9; ++i) k.Pinf[i] = variance * P[i];
  return k;
}
__device__ __forceinline__ void disc_A(const Kern& k, float dt, float* A) {
  float e = expf(-k.lam * dt);
  float h = 0.5f * dt * dt;
#pragma unroll
  for (int i = 0; i < 9; ++i) {
    float idv = (i == 0 || i == 4 || i == 8) ? 1.0f : 0.0f;
    A[i] = e * (idv + dt * k.Nm[i] + h * k.Nm2[i]);
  }
}
__device__ __forceinline__ void disc_AQ(const Kern& k, float dt, float* A, float* Q) {
  disc_A(k, dt, A);
  float T1[9]; mm3(A, k.Pinf, T1);
  float T2[9]; mm3nt(T1, A, T2);
#pragma unroll
  for (int i = 0; i < 9; ++i) Q[i] = k.Pinf[i] - T2[i];
}

// ------------------------ binary searches (stable merge) -------------------
__device__ __forceinline__ int lb_cnt(const float* a, int n, float x) { // #(a[k] <  x)
  int lo = 0, hi = n;
  while (lo < hi) { int mid = (lo + hi) >> 1; if (a[mid] < x) lo = mid + 1; else hi = mid; }
  return lo;
}
__device__ __forceinline__ int ub_cnt(const float* a, int n, float x) { // #(a[k] <= x)
  int lo = 0, hi = n;
  while (lo < hi) { int mid = (lo + hi) >> 1; if (a[mid] <= x) lo = mid + 1; else hi = mid; }
  return lo;
}

// --------------------- TDM staging of tm tile into LDS ---------------------
// 1D tile of TILE_ELEMS f32, +1 DWORD pad per 64 DWORDs -> LDS stride 195 per
// chunk (bank stride 3, conflict free).  Issued by wave 0 only; TDM ignores
// EXEC and uses TENSORcnt.
__device__ __forceinline__ float tile_tm(const float* s, int g) { return s[g + (g >> 6)]; }

__device__ __forceinline__ void tdm_stage_tile(const float* gsrc) {
  if (threadIdx.x == 0) {
    unsigned long long ga = (unsigned long long)gsrc;
    u32x4 g0;
    g0[0] = 1u;                                   // count=1 (valid descriptor)
    g0[1] = 0u;                                   // lds_addr: dynamic LDS base (no static LDS here)
    g0[2] = (unsigned)ga;                         // global_addr[31:0]
    g0[3] = (unsigned)((ga >> 32) & 0x1FFFFFFull) // global_addr[56:32]
          | (2u << 30);                           // type=2 ("image")
    i32x8 g1;
    g1[0] = (2 << 16)      // data_size = 4 bytes
          | (1 << 20)      // pad_enable
          | (5 << 22)      // pad_interval = 64 DWORDs
          | (0 << 25);     // pad_amount = 1 DWORD
    g1[1] = (int)((unsigned)(TILE_ELEMS & 0xFFFF) << 16);  // tensor_dim0[15:0] @ bits63:48
    g1[2] = (int)(((TILE_ELEMS >> 16) & 0xFFFF) | (1u << 16)); // tensor_dim0 hi | tensor_dim1=1
    g1[3] = (int)((unsigned)(TILE_ELEMS & 0xFFFF) << 16);  // tile_dim0 @ bits127:112
    g1[4] = 1;             // tile_dim1 = 1, tile_dim2 = 0
    g1[5] = TILE_ELEMS;    // tensor_dim0_stride low32
    g1[6] = 0;
    g1[7] = 0;
    i32x4 gz; gz[0] = 0; gz[1] = 0; gz[2] = 0; gz[3] = 0;
    i32x8 gz8; gz8[0]=0; gz8[1]=0; gz8[2]=0; gz8[3]=0; gz8[4]=0; gz8[5]=0; gz8[6]=0; gz8[7]=0;
    // amdgpu-toolchain (clang-23 / therock-10.0) 6-arg form
    __builtin_amdgcn_tensor_load_to_lds(g0, g1, gz, gz, gz8, 0);
    __builtin_amdgcn_s_wait_tensorcnt(0);
  }
  __syncthreads();
}

// ---------------------- filter scan element (SGF 5-tuple) ------------------
struct FAgg { float A[9]; float b[3]; float C[9]; float eta[3]; float J[9]; };

__device__ __forceinline__ void fagg_ident(FAgg& s) {
#pragma unroll
  for (int i = 0; i < 9; ++i) { s.A[i] = 0.f; s.C[i] = 0.f; s.J[i] = 0.f; }
#pragma unroll
  for (int i = 0; i < 3; ++i) { s.b[i] = 0.f; s.eta[i] = 0.f; }
  s.A[0] = s.A[4] = s.A[8] = 1.f;
}
__device__ __forceinline__ void fagg_store(float* d, const FAgg& s) {
#pragma unroll
  for (int q = 0; q < 9; ++q) d[q] = s.A[q];
#pragma unroll
  for (int q = 0; q < 3; ++q) d[9+q] = s.b[q];
#pragma unroll
  for (int q = 0; q < 9; ++q) d[12+q] = s.C[q];
#pragma unroll
  for (int q = 0; q < 3; ++q) d[21+q] = s.eta[q];
#pragma unroll
  for (int q = 0; q < 9; ++q) d[24+q] = s.J[q];
}
__device__ __forceinline__ void fagg_load(FAgg& s, const float* d) {
#pragma unroll
  for (int q = 0; q < 9; ++q) s.A[q] = d[q];
#pragma unroll
  for (int q = 0; q < 3; ++q) s.b[q] = d[9+q];
#pragma unroll
  for (int q = 0; q < 9; ++q) s.C[q] = d[12+q];
#pragma unroll
  for (int q = 0; q < 3; ++q) s.eta[q] = d[21+q];
#pragma unroll
  for (int q = 0; q < 9; ++q) s.J[q] = d[24+q];
}

// fold one per-step leaf into acc (rank-1 obs -> Sherman-Morrison, no inv3)
__device__ __forceinline__ void fagg_step(FAgg& s, const float* A, const float* Q,
                                          float y, float R, bool obs) {
  if (!obs) {   // leaf = (A,0,Q,0,0): pure affine propagation
    float nA[9]; mm3(A, s.A, nA);
    float nb[3]; mv3(A, s.b, nb);
    float T[9]; mm3(A, s.C, T);
    float nC[9]; mm3nt(T, A, nC);
#pragma unroll
    for (int q = 0; q < 9; ++q) { s.A[q] = nA[q]; s.C[q] = nC[q] + Q[q]; }
#pragma unroll
    for (int q = 0; q < 3; ++q) s.b[q] = nb[q];
    return;
  }
  float S  = Q[0] + R;
  float iS = 1.0f / S;
  float a0[3] = {A[0], A[1], A[2]};                  // first row of transition
  float K[3]  = {Q[0]*iS, Q[3]*iS, Q[6]*iS};
  float u[3]; mv3(s.C, a0, u);
#pragma unroll
  for (int q = 0; q < 3; ++q) u[q] *= iS;            // u = C_acc a0 / S
  float d = 1.0f + a0[0]*u[0] + a0[1]*u[1] + a0[2]*u[2];
  float idd = 1.0f / d;
  float w[3];                                        // w = A_acc^T a0
#pragma unroll
  for (int j = 0; j < 3; ++j) w[j] = s.A[0*3+j]*a0[0] + s.A[1*3+j]*a0[1] + s.A[2*3+j]*a0[2];
  float Al[9], M1[9];
#pragma unroll
  for (int i = 0; i < 3; ++i)
#pragma unroll
    for (int j = 0; j < 3; ++j) {
      Al[i*3+j] = A[i*3+j] - K[i]*a0[j];
      M1[i*3+j] = s.A[i*3+j] - u[i]*w[j]*idd;
    }
  float nA[9]; mm3(Al, M1, nA);
  float tv[3];
#pragma unroll
  for (int q = 0; q < 3; ++q) tv[q] = s.b[q] + u[q]*y;
  float z = a0[0]*tv[0] + a0[1]*tv[1] + a0[2]*tv[2];
  float tv2[3];
#pragma unroll
  for (int q = 0; q < 3; ++q) tv2[q] = tv[q] - u[q]*z*idd;
  float nb[3]; mv3(Al, tv2, nb);
#pragma unroll
  for (int q = 0; q < 3; ++q) nb[q] += K[q]*y;
  float Sd = S * idd;
  float M2[9];
#pragma unroll
  for (int i = 0; i < 3; ++i)
#pragma unroll
    for (int j = 0; j < 3; ++j) M2[i*3+j] = s.C[i*3+j] - Sd*u[i]*u[j];
  float T2[9]; mm3(Al, M2, T2);
  float nC[9]; mm3nt(T2, Al, nC);
#pragma unroll
  for (int i = 0; i < 3; ++i)
#pragma unroll
    for (int j = 0; j < 3; ++j) nC[i*3+j] += Q[i*3+j] - K[i]*Q[j];   // + C_leaf
  float s1 = (y - (a0[0]*s.b[0] + a0[1]*s.b[1] + a0[2]*s.b[2])) * iS;
  float f1 = s1 * idd, f2 = iS * idd;
#pragma unroll
  for (int q = 0; q < 3; ++q) s.eta[q] += w[q]*f1;
#pragma unroll
  for (int i = 0; i < 3; ++i)
#pragma unroll
    for (int j = 0; j < 3; ++j) s.J[i*3+j] += w[i]*w[j]*f2;
#pragma unroll
  for (int q = 0; q < 9; ++q) { s.A[q] = nA[q]; s.C[q] = nC[q]; }
#pragma unroll
  for (int q = 0; q < 3; ++q) s.b[q] = nb[q];
}

// general combine o = x (earlier) (x) y (later)   -- used in the chunk scan
__device__ __forceinline__ void fagg_combine(FAgg& o, const FAgg& x, const FAgg& y) {
  float M1[9]; mm3(x.C, y.J, M1); M1[0] += 1.f; M1[4] += 1.f; M1[8] += 1.f;
  float Z1[9]; inv3(M1, Z1);
  float TA[9]; mm3(y.A, Z1, TA);
  mm3(TA, x.A, o.A);
  float tb[3]; mv3(x.C, y.eta, tb);
#pragma unroll
  for (int q = 0; q < 3; ++q) tb[q] += x.b[q];
  float tb2[3]; mv3(TA, tb, tb2);
#pragma unroll
  for (int q = 0; q < 3; ++q) o.b[q] = tb2[q] + y.b[q];
  float TC[9]; mm3(TA, x.C, TC);
  float TC2[9]; mm3nt(TC, y.A, TC2);
#pragma unroll
  for (int q = 0; q < 9; ++q) o.C[q] = TC2[q] + y.C[q];
  float M2[9]; mm3(y.J, x.C, M2); M2[0] += 1.f; M2[4] += 1.f; M2[8] += 1.f;
  float Z2[9]; inv3(M2, Z2);
  float TT[9]; mm3tn(x.A, Z2, TT);                 // A_x^T Z2
  float te[3]; mv3(y.J, x.b, te);
#pragma unroll
  for (int q = 0; q < 3; ++q) te[q] = y.eta[q] - te[q];
  float te2[3]; mv3(TT, te, te2);
#pragma unroll
  for (int q = 0; q < 3; ++q) o.eta[q] = te2[q] + x.eta[q];
  float TJ[9]; mm3(TT, y.J, TJ);
  float TJ2[9]; mm3(TJ, x.A, TJ2);
#pragma unroll
  for (int q = 0; q < 9; ++q) o.J[q] = TJ2[q] + x.J[q];
}

// --------------------------- smoother elements -----------------------------
struct SElem { float E[9]; float g[3]; float L[9]; };

__device__ __forceinline__ void build_selem(int t, const float* ws, const Kern& k, SElem& e) {
  const float* r = ws + OFF_REC + (size_t)t * 24;   // [m_pred3, P_pred9, m_f3, P_f9]
  if (t == T_TOT - 1) {
#pragma unroll
    for (int q = 0; q < 9; ++q) { e.E[q] = 0.f; e.L[q] = r[15+q]; }
#pragma unroll
    for (int q = 0; q < 3; ++q) e.g[q] = r[12+q];
  } else {
    const float* rn = r + 24;
    const float* tm = ws + OFF_TM;
    float dtn = tm[t+1] - tm[t];
    float A1[9]; disc_A(k, dtn, A1);
    float Pi[9]; inv3(rn + 3, Pi);                  // P_pred_{t+1}^{-1}
    float T1[9]; mm3nt(r + 15, A1, T1);             // P_f A1^T
    mm3(T1, Pi, e.E);                               // G
    float Gm[3]; mv3(e.E, rn + 0, Gm);
#pragma unroll
    for (int q = 0; q < 3; ++q) e.g[q] = r[12+q] - Gm[q];
    float T2[9]; mm3(e.E, rn + 3, T2);
    float T3[9]; mm3nt(T2, e.E, T3);
#pragma unroll
    for (int q = 0; q < 9; ++q) e.L[q] = r[15+q] - T3[q];
  }
}
__device__ __forceinline__ void selem_combine(SElem& o, const SElem& x, const SElem& y) {
  float T1[9]; mm3(x.E, y.L, T1);
  float T2[9]; mm3nt(T1, x.E, T2);
#pragma unroll
  for (int q = 0; q < 9; ++q) o.L[q] = T2[q] + x.L[q];
  float gv[3]; mv3(x.E, y.g, gv);
#pragma unroll
  for (int q = 0; q < 3; ++q) o.g[q] = gv[q] + x.g[q];
  mm3(x.E, y.E, o.E);
}

// LDS element layout for smoother scan: [E 0..8][g 9..11][L 12..20], stride 21.
// homogeneous 4x4 view of (E,g):
__device__ __forceinline__ float homX(const float* buf, int e, int r, int c) {
  if (r == 3) return (c == 3) ? 1.0f : 0.0f;
  return (c == 3) ? buf[e*21 + 9 + r] : buf[e*21 + r*3 + c];
}

// ------------------------------- kernels -----------------------------------
__global__ void k_merge_train(const float* __restrict__ times, const float* __restrict__ tstar,
                              const float* __restrict__ sn1, const float* __restrict__ sn2,
                              const float* __restrict__ pmean, float* __restrict__ ws) {
  int i = blockIdx.x * blockDim.x + threadIdx.x;
  if (i >= N_TRAIN) return;
  float x = times[i];
  int pos = i + lb_cnt(tstar, M_TEST, x);           // stable: train before test on ties
  float s2 = sn2[i];
  ws[OFF_TM + pos]  = x;
  ws[OFF_RES + pos] = sn1[i] / s2 - *pmean;
  ws[OFF_R + pos]   = 1.0f / s2;
  ((int*)(ws + OFF_TIDX))[pos] = -1;
}

__global__ void k_merge_test(const float* __restrict__ times, const float* __restrict__ tstar,
                             float* __restrict__ ws) {
  int j = blockIdx.x * blockDim.x + threadIdx.x;
  if (j >= M_TEST) return;
  float x = tstar[j];
  int pos = j + ub_cnt(times, N_TRAIN, x);
  ws[OFF_TM + pos]  = x;
  ws[OFF_RES + pos] = 0.0f;
  ws[OFF_R + pos]   = 1.0f;
  ((int*)(ws + OFF_TIDX))[pos] = j;
}

__global__ void __launch_bounds__(256) k_filter_chunks(float* __restrict__ ws,
                                                       const float* pvar, const float* plen) {
  extern __shared__ float smA[];
  const float* tm = ws + OFF_TM;
  int c  = blockIdx.x * blockDim.x + threadIdx.x;   // chunk id
  int cb = blockIdx.x * CPB;
  tdm_stage_tile(tm + (size_t)cb * CHLEN);          // TDM: tm tile -> LDS (padded)
  Kern k = make_kern(*pvar, *plen);
  const float* resid = ws + OFF_RES;
  const float* Rf    = ws + OFF_R;
  const int* tidx    = (const int*)(ws + OFF_TIDX);
  FAgg s; fagg_ident(s);
  float tprev;
  if (c == 0)       tprev = 0.0f;
  else if (c == cb) tprev = tm[(size_t)c * CHLEN - 1];
  else              tprev = tile_tm(smA, (c - cb) * CHLEN - 1);
  for (int ss = 0; ss < CHLEN; ++ss) {
    int t = c * CHLEN + ss;
    float tc = tile_tm(smA, (c - cb) * CHLEN + ss);
    float dt = (t == 0) ? 0.0f : (tc - tprev);
    tprev = tc;
    float A[9], Q[9]; disc_AQ(k, dt, A, Q);
    fagg_step(s, A, Q, resid[t], Rf[t], tidx[t] < 0);
  }
  fagg_store(ws + OFF_FAGG + (size_t)c * 33, s);
}

__global__ void __launch_bounds__(1024) k_filter_scan(float* __restrict__ ws,
                                                      const float* pvar, const float* plen) {
  extern __shared__ float smB[];                    // NCHUNK x 33 floats (135 KB)
  int i = threadIdx.x;
  FAgg v; fagg_load(v, ws + OFF_FAGG + (size_t)i * 33);
  fagg_store(smB + (size_t)i * 33, v);
  __syncthreads();
  for (int d = 1; d < NCHUNK; d <<= 1) {            // Kogge-Stone inclusive prefix
    FAgg p;
    if (i >= d) fagg_load(p, smB + (size_t)(i - d) * 33);
    __syncthreads();
    if (i >= d) { FAgg nv; fagg_combine(nv, p, v); v = nv; fagg_store(smB + (size_t)i * 33, v); }
    __syncthreads();
  }
  // incoming filter state for chunk i = apply(prior (x) prefix_{i-1})
  Kern k = make_kern(*pvar, *plen);
  float m[3], P[9];
  if (i == 0) {
#pragma unroll
    for (int q = 0; q < 3; ++q) m[q] = 0.f;
#pragma unroll
    for (int q = 0; q < 9; ++q) P[q] = k.Pinf[q];
  } else {
    const float* g = smB + (size_t)(i - 1) * 33;    // A|b|C|eta|J
    float M1[9]; mm3(k.Pinf, g + 24, M1); M1[0] += 1.f; M1[4] += 1.f; M1[8] += 1.f;
    float Z[9]; inv3(M1, Z);
    float TA[9]; mm3(g + 0, Z, TA);
    float pe[3]; mv3(k.Pinf, g + 21, pe);
    mv3(TA, pe, m);
#pragma unroll
    for (int q = 0; q < 3; ++q) m[q] += g[9+q];
    float T2[9]; mm3(TA, k.Pinf, T2);
    float T3[9]; mm3nt(T2, g + 0, T3);
#pragma unroll
    for (int q = 0; q < 9; ++q) P[q] = T3[q] + g[12+q];
  }
  float* fin = ws + OFF_FIN + (size_t)i * 12;
#pragma unroll
  for (int q = 0; q < 3; ++q) fin[q] = m[q];
#pragma unroll
  for (int q = 0; q < 9; ++q) fin[3+q] = P[q];
}

__global__ void __launch_bounds__(256) k_filter_sweep(float* __restrict__ ws,
                                                      const float* pvar, const float* plen) {
  extern __shared__ float smD[];
  const float* tm = ws + OFF_TM;
  int c  = blockIdx.x * blockDim.x + threadIdx.x;
  int cb = blockIdx.x * CPB;
  tdm_stage_tile(tm + (size_t)cb * CHLEN);
  Kern k = make_kern(*pvar, *plen);
  const float* fin = ws + OFF_FIN + (size_t)c * 12;
  float m[3] = {fin[0], fin[1], fin[2]};
  float P[9];
#pragma unroll
  for (int q = 0; q < 9; ++q) P[q] = fin[3+q];
  const float* resid = ws + OFF_RES;
  const float* Rf    = ws + OFF_R;
  const int* tidx    = (const int*)(ws + OFF_TIDX);
  float* rec = ws + OFF_REC;
  float tprev;
  if (c == 0)       tprev = 0.0f;
  else if (c == cb) tprev = tm[(size_t)c * CHLEN - 1];
  else              tprev = tile_tm(smD, (c - cb) * CHLEN - 1);
  for (int ss = 0; ss < CHLEN; ++ss) {
    int t = c * CHLEN + ss;
    float tc = tile_tm(smD, (c - cb) * CHLEN + ss);
    float dt = (t == 0) ? 0.0f : (tc - tprev);
    tprev = tc;
    float A[9], Q[9]; disc_AQ(k, dt, A, Q);
    float mp[3]; mv3(A, m, mp);
    float T[9]; mm3(A, P, T);
    float Pp[9]; mm3nt(T, A, Pp);
#pragma unroll
    for (int q = 0; q < 9; ++q) Pp[q] += Q[q];
    float* r = rec + (size_t)t * 24;
#pragma unroll
    for (int q = 0; q < 3; ++q) r[q] = mp[q];
#pragma unroll
    for (int q = 0; q < 9; ++q) r[3+q] = Pp[q];
    float S = Pp[0] + Rf[t];
    if (tidx[t] < 0) {
      float iS = 1.0f / S;
      float K[3] = {Pp[0]*iS, Pp[3]*iS, Pp[6]*iS};
      float v = resid[t] - mp[0];
#pragma unroll
      for (int q = 0; q < 3; ++q) m[q] = mp[q] + K[q]*v;
#pragma unroll
      for (int i2 = 0; i2 < 3; ++i2)
#pragma unroll
        for (int j2 = 0; j2 < 3; ++j2) P[i2*3+j2] = Pp[i2*3+j2] - K[i2]*K[j2]*S;
    } else {
#pragma unroll
      for (int q = 0; q < 3; ++q) m[q] = mp[q];
#pragma unroll
      for (int q = 0; q < 9; ++q) P[q] = Pp[q];
    }
#pragma unroll
    for (int q = 0; q < 3; ++q) r[12+q] = m[q];
#pragma unroll
    for (int q = 0; q < 9; ++q) r[15+q] = P[q];
  }
}

__global__ void __launch_bounds__(256) k_smooth_chunks(float* __restrict__ ws,
                                                       const float* pvar, const float* plen) {
  int c = blockIdx.x * blockDim.x + threadIdx.x;
  Kern k = make_kern(*pvar, *plen);
  SElem acc;
#pragma unroll
  for (int q = 0; q < 9; ++q) { acc.E[q] = 0.f; acc.L[q] = 0.f; }
#pragma unroll
  for (int q = 0; q < 3; ++q) acc.g[q] = 0.f;
  acc.E[0] = acc.E[4] = acc.E[8] = 1.f;
  for (int ss = 0; ss < CHLEN; ++ss) {
    int t = c * CHLEN + ss;
    SElem e; build_selem(t, ws, k, e);
    SElem nx; selem_combine(nx, acc, e);
    acc = nx;
  }
  float* d = ws + OFF_SAGG + (size_t)c * 21;
#pragma unroll
  for (int q = 0; q < 9; ++q) d[q] = acc.E[q];
#pragma unroll
  for (int q = 0; q < 3; ++q) d[9+q] = acc.g[q];
#pragma unroll
  for (int q = 0; q < 9; ++q) d[12+q] = acc.L[q];
}

// Backward (suffix) scan over chunk aggregates.  (E,g) part = homogeneous 4x4
// products batched 4-per-WMMA (v_wmma_f32_16x16x4_f32, K=4); L part per-thread.
__global__ void __launch_bounds__(1024) k_smooth_scan(float* __restrict__ ws) {
  extern __shared__ float smC[];
  float* b0 = smC;
  float* b1 = smC + (size_t)NCHUNK * 21;
  int i = threadIdx.x;
  const float* sa = ws + OFF_SAGG;
#pragma unroll
  for (int q = 0; q < 21; ++q) b0[i*21+q] = sa[(size_t)i*21+q];
  __syncthreads();
  float* bc = b0; float* bn = b1;
  int lane = i & 31;
  int wavebase = i & ~31;
  int n  = lane & 15;
  int p  = n >> 2;                 // which of the 4 stacked 4x4 blocks
  int rc = n & 3;                  // row (A) / col (B) inside the block
  int k0 = (lane < 16) ? 0 : 2;    // K columns this half-wave supplies
  bool hi = lane >= 16;
  bool useful = ((n < 8) == (!hi));// lanes holding diagonal blocks of D
  for (int d = 1; d < NCHUNK; d <<= 1) {
    { // ---- per-thread L part: L' = E_i L_{i+d} E_i^T + L_i ----
      float E[9], L[9], Ly[9];
#pragma unroll
      for (int q = 0; q < 9; ++q) { E[q] = bc[i*21+q]; L[q] = bc[i*21+12+q]; }
      int j = i + d;
      if (j < NCHUNK) {
#pragma unroll
        for (int q = 0; q < 9; ++q) Ly[q] = bc[j*21+12+q];
      } else {
#pragma unroll
        for (int q = 0; q < 9; ++q) Ly[q] = 0.f;
      }
      float T1[9]; mm3(E, Ly, T1);
      float T2[9]; mm3nt(T1, E, T2);
#pragma unroll
      for (int q = 0; q < 9; ++q) bn[i*21+12+q] = T2[q] + L[q];
    }
    // ---- wave-cooperative (E,g): 8 WMMAs cover the wave's 32 elements ----
    for (int q8 = 0; q8 < 8; ++q8) {
      int eA = wavebase + (q8 << 2) + p;           // left (own) element
      v2f av, bv;
      av.x = homX(bc, eA, rc, k0);
      av.y = homX(bc, eA, rc, k0 + 1);
      int eB = eA + d;                             // right (partner) or identity
      if (eB < NCHUNK) {
        bv.x = homX(bc, eB, k0,     rc);
        bv.y = homX(bc, eB, k0 + 1, rc);
      } else {
        bv.x = (k0 == rc)       ? 1.f : 0.f;
        bv.y = ((k0 + 1) == rc) ? 1.f : 0.f;
      }
      v8f cz = {0.f,0.f,0.f,0.f,0.f,0.f,0.f,0.f};
      v8f dd = __builtin_amdgcn_wmma_f32_16x16x4_f32(false, av, false, bv,
                                                     (short)0, cz, false, false);
      if (useful) {
#pragma unroll
        for (int z = 0; z < 8; ++z) {
          int Mrow = z + (hi ? 8 : 0);
          int i2 = Mrow - 4 * p;                   // row inside diagonal block
          if (i2 >= 0 && i2 < 3) {
            float val = dd[z];
            if (rc < 3) bn[eA*21 + i2*3 + rc] = val;   // E'
            else        bn[eA*21 + 9 + i2]   = val;    // g'
          }
        }
      }
    }
    __syncthreads();
    float* tsw = bc; bc = bn; bn = tsw;
  }
  // incoming smoothed state at chunk c's right boundary = suffix_{c+1}.(g,L)
  float* si = ws + OFF_SIN + (size_t)i * 12;
  if (i == NCHUNK - 1) {
#pragma unroll
    for (int q = 0; q < 12; ++q) si[q] = 0.f;      // killed by E=0 of last leaf
  } else {
#pragma unroll
    for (int q = 0; q < 3; ++q) si[q] = bc[(i+1)*21 + 9 + q];
#pragma unroll
    for (int q = 0; q < 9; ++q) si[3+q] = bc[(i+1)*21 + 12 + q];
  }
}

__global__ void __launch_bounds__(256) k_smooth_sweep(float* __restrict__ ws,
                                                      const float* pvar, const float* plen,
                                                      const float* pmean, float* __restrict__ out) {
  int c = blockIdx.x * blockDim.x + threadIdx.x;
  Kern k = make_kern(*pvar, *plen);
  const float* si = ws + OFF_SIN + (size_t)c * 12;
  float nm[3] = {si[0], si[1], si[2]};
  float nP[9];
#pragma unroll
  for (int q = 0; q < 9; ++q) nP[q] = si[3+q];
  const int* tidx = (const int*)(ws + OFF_TIDX);
  float mc = *pmean;
  for (int ss = CHLEN - 1; ss >= 0; --ss) {
    int t = c * CHLEN + ss;
    SElem e; build_selem(t, ws, k, e);
    float m[3]; mv3(e.E, nm, m);
#pragma unroll
    for (int q = 0; q < 3; ++q) m[q] += e.g[q];
    float T[9]; mm3(e.E, nP, T);
    float P[9]; mm3nt(T, e.E, P);
#pragma unroll
    for (int q = 0; q < 9; ++q) P[q] += e.L[q];
    int ti = tidx[t];
    if (ti >= 0) {
      out[ti] = m[0] + mc;
      out[M_TEST + ti] = fmaxf(P[0], 0.0f);
    }
#pragma unroll
    for (int q = 0; q < 3; ++q) nm[q] = m[q];
#pragma unroll
    for (int q = 0; q < 9; ++q) nP[q] = P[q];
  }
}

// ------------------------------ launcher -----------------------------------
extern "C" void kernel_launch(void* const* d_in, const int* in_sizes, int n_in,
                              void* d_out, int out_size, void* d_ws, size_t ws_size,
                              hipStream_t stream) {
  (void)in_sizes; (void)n_in; (void)out_size; (void)ws_size;
  const float* times = (const float*)d_in[0];
  const float* tstar = (const float*)d_in[1];
  const float* sn1   = (const float*)d_in[2];
  const float* sn2   = (const float*)d_in[3];
  const float* pvar  = (const float*)d_in[4];
  const float* plen  = (const float*)d_in[5];
  const float* pmean = (const float*)d_in[6];
  float* ws  = (float*)d_ws;
  float* out = (float*)d_out;

  dim3 b256(256);
  size_t tileB = (size_t)TILE_LDS_DW * 4;                 // 199,680 B (<320 KB/WGP)

  k_merge_train<<<N_TRAIN/256, b256, 0, stream>>>(times, tstar, sn1, sn2, pmean, ws);
  k_merge_test <<<M_TEST/256,  b256, 0, stream>>>(times, tstar, ws);
  k_filter_chunks<<<NCHUNK/CPB, b256, tileB, stream>>>(ws, pvar, plen);
  k_filter_scan  <<<1, 1024, (size_t)NCHUNK*33*4, stream>>>(ws, pvar, plen);
  k_filter_sweep <<<NCHUNK/CPB, b256, tileB, stream>>>(ws, pvar, plen);
  k_smooth_chunks<<<NCHUNK/CPB, b256, 0, stream>>>(ws, pvar, plen);
  k_smooth_scan  <<<1, 1024, (size_t)2*NCHUNK*21*4, stream>>>(ws);
  k_smooth_sweep <<<NCHUNK/CPB, b256, 0, stream>>>(ws, pvar, plen, pmean, out);
}